// OrderPredictorPCT_5377299054753
// MI455X (gfx1250) — compile-verified
//
#include <hip/hip_runtime.h>
#include <hip/hip_bf16.h>
#include <stdint.h>

// ---------------------------------------------------------------------------
// OrderPredictorPCT forward for MI455X (gfx1250, wave32).
// GEMMs use V_WMMA_F32_16X16X4_F32 (full fp32 precision on the matrix pipe);
// weight tiles are staged into LDS with the Tensor Data Mover
// (tensor_load_to_lds + s_wait_tensorcnt), double-buffered so the DMA of
// tile t+1 overlaps the WMMA stream of tile t (TENSORcnt is in-order per
// wave, so waiting tensorcnt<=1 guarantees the older tile landed). Small ops
// (attention S=28, LayerNorm, FF=16, Sinkhorn 12x12) run on VALU with
// wave32 shuffles.
// ---------------------------------------------------------------------------

typedef __attribute__((ext_vector_type(2)))  float    v2f;
typedef __attribute__((ext_vector_type(8)))  float    v8f;
typedef __attribute__((ext_vector_type(4)))  unsigned v4u;
typedef __attribute__((ext_vector_type(4)))  int      v4i;
typedef __attribute__((ext_vector_type(8)))  int      v8i;

#define B_    512
#define KOBJ  12
#define WSENT 16
#define NPTS  128
#define DMODEL 256
#define NHEAD 8
#define HDIM  32
#define FFDIM 16
#define NLAYER 8
#define SLEN  28          // WSENT + KOBJ
#define NSAMP 5
#define NITER 20
#define BK    (B_*KOBJ)   // 6144
#define ROWS  (B_*SLEN)   // 14336

// ---------------------------------------------------------------------------
// TDM 2D tile load: DMA (tile 64 rows x 64 cols of f32, row stride =
// row_stride_elems) from global into LDS at lds_off. tensor_dim set to the
// *remaining* extent so out-of-bounds elements are zero-filled by the TDM.
// D# bitfields per CDNA5 ISA ch.10.7 / 8.3-8.4. This toolchain exposes the
// 6-arg builtin: (g0 v4u, g1 v8i, g2 v4i, g3 v4i, ext v8i, cpol); the extra
// group is zero-filled (2D tensor needs only groups 0-1).
// ---------------------------------------------------------------------------
__device__ __forceinline__ void tdm_load_tile_64x64(
    unsigned lds_off, const float* gptr, int rem_cols, int rem_rows,
    int row_stride_elems) {
  unsigned long long ga = (unsigned long long)(uintptr_t)gptr;
  v4u g0;
  g0[0] = 1u;                                      // count=1, user mode
  g0[1] = lds_off;                                 // lds_addr (bytes)
  g0[2] = (unsigned)(ga & 0xffffffffu);            // global_addr[31:0]
  g0[3] = (unsigned)((ga >> 32) & 0x01ffffffu)     // global_addr[56:32]
        | (2u << 30);                              // type = 2 (image)
  unsigned td0 = (unsigned)rem_cols;
  unsigned td1 = (unsigned)rem_rows;
  unsigned st0 = (unsigned)row_stride_elems;
  v8i g1;
  g1[0] = (int)(2u << 16);                         // wg_mask=0, data_size=4B
  g1[1] = (int)((td0 & 0xffffu) << 16);            // barrier_addr=0 | td0.lo16
  g1[2] = (int)(((td0 >> 16) & 0xffffu) | ((td1 & 0xffffu) << 16));
  g1[3] = (int)(((td1 >> 16) & 0xffffu) | (64u << 16));   // tile_dim0 = 64
  g1[4] = (int)(64u);                              // tile_dim1=64, tile_dim2=0
  g1[5] = (int)st0;                                // tensor_dim0_stride[31:0]
  g1[6] = 0;                                       // stride0.hi16 | stride1.lo16
  g1[7] = 0;                                       // stride1.hi32
  v4i g2 = {0, 0, 0, 0};
  v4i g3 = {0, 0, 0, 0};
  v8i g4 = {0, 0, 0, 0, 0, 0, 0, 0};
  __builtin_amdgcn_tensor_load_to_lds(g0, g1, g2, g3, g4, 0);
}

// ---------------------------------------------------------------------------
// Generic fp32 WMMA GEMM: C[M,N] = A[M,Kd] @ W[Kd,N] + bias (+relu).
// Block = 128 threads (4 waves). Block tile: 64(M) x 64(N); wave w owns rows
// [64*by + 16*w, +16). K tiled by 64 via double-buffered TDM into LDS.
// M % 64 == 0, Kd % 4 == 0; N ragged (guarded stores, TDM zero-fill).
// ---------------------------------------------------------------------------
template <bool RELU>
__global__ void __launch_bounds__(128) gemm_bias_kernel(
    const float* __restrict__ A, const float* __restrict__ Wt,
    const float* __restrict__ bias, float* __restrict__ C,
    int M, int Kd, int Nout) {
  __shared__ float wtile[2][64 * 64];
  const int tid    = threadIdx.x;
  const int wave   = tid >> 5;
  const int lane   = tid & 31;
  const int lane16 = lane & 15;
  const int khalf  = lane >> 4;          // 0: K pair {0,1}, 1: K pair {2,3}
  const int n0     = blockIdx.x * 64;
  const int m0     = blockIdx.y * 64 + wave * 16;

  // Visible LDS stores so the compiler cannot assume wtile is never written
  // (TDM writes it behind alias analysis' back); also defines the buffers.
  for (int i = tid; i < 2 * 64 * 64; i += 128) wtile[0][i & 4095] = 0.f,
                                               wtile[i >> 12][i & 4095] = 0.f;
  __syncthreads();

  v8f cacc[4];
  #pragma unroll
  for (int j = 0; j < 4; ++j) cacc[j] = (v8f){0,0,0,0,0,0,0,0};

  const unsigned lds0 = (unsigned)(uintptr_t)&wtile[0][0];
  const unsigned lds1 = (unsigned)(uintptr_t)&wtile[1][0];
  const float* arow = A + (size_t)(m0 + lane16) * Kd + 2 * khalf;
  const int ktiles = (Kd + 63) >> 6;

  if (wave == 0)                          // prefetch tile 0
    tdm_load_tile_64x64(lds0, Wt + n0, Nout - n0, Kd, Nout);

  for (int t = 0; t < ktiles; ++t) {
    const int k0 = t * 64;
    if (wave == 0) {
      if (t + 1 < ktiles) {               // prefetch tile t+1 into other buf
        tdm_load_tile_64x64((t & 1) ? lds0 : lds1,
                            Wt + (size_t)(k0 + 64) * Nout + n0,
                            Nout - n0, Kd - (k0 + 64), Nout);
        __builtin_amdgcn_s_wait_tensorcnt(1);   // oldest (tile t) landed
      } else {
        __builtin_amdgcn_s_wait_tensorcnt(0);
      }
    }
    __syncthreads();                      // tile t visible to all waves

    const float* wbuf = wtile[t & 1];
    const int kt = (Kd - k0) < 64 ? (Kd - k0) : 64;
    #pragma unroll 4
    for (int kk = 0; kk < kt; kk += 4) {
      // A fragment (16x4): lane m=lane16, holds K = kk + 2*khalf + {0,1}
      v2f a = *(const v2f*)(arow + k0 + kk);
      const float* wl = wbuf + (kk + 2 * khalf) * 64 + lane16;
      #pragma unroll
      for (int j = 0; j < 4; ++j) {
        v2f b;
        b.x = wl[j * 16];                 // B row (kk+2*khalf)   col 16j+lane16
        b.y = wl[j * 16 + 64];            // B row (kk+2*khalf+1)
        cacc[j] = __builtin_amdgcn_wmma_f32_16x16x4_f32(
            false, a, false, b, (short)0, cacc[j], false, false);
      }
    }
    __syncthreads();                      // tile t consumed; buf reusable
  }

  // Store: C/D layout -> vgpr v: rows m0 + v + 8*khalf, col n0+16j+lane16.
  #pragma unroll
  for (int j = 0; j < 4; ++j) {
    const int col = n0 + j * 16 + lane16;
    if (col < Nout) {
      const float bv = bias ? bias[col] : 0.f;
      #pragma unroll
      for (int v = 0; v < 8; ++v) {
        float val = cacc[j][v] + bv;
        if (RELU) val = fmaxf(val, 0.f);
        C[(size_t)(m0 + v + 8 * khalf) * Nout + col] = val;
      }
    }
  }
}

// ---------------------------------------------------------------------------
// Point encoder: one block per object (128 threads = 128 points).
// pts = [xyz - mean(xyz), rgb]; relu(pts@pW1+pb1); relu(@pW2+pb2); max over
// points (LDS int atomicMax -- valid: post-ReLU values are >= 0).
// ---------------------------------------------------------------------------
__global__ void __launch_bounds__(128) point_enc_kernel(
    const float* __restrict__ xyzs, const float* __restrict__ rgbs,
    const float* __restrict__ pW1, const float* __restrict__ pb1,
    const float* __restrict__ pW2, const float* __restrict__ pb2,
    float* __restrict__ pooled) {
  __shared__ float sW1[6 * 64];
  __shared__ float sB1[64];
  __shared__ float sW2[64 * 128];
  __shared__ float sB2[128];
  __shared__ float cent[3];
  __shared__ float pool[128];

  const int obj = blockIdx.x;
  const int tid = threadIdx.x;

  for (int i = tid; i < 6 * 64; i += 128)   sW1[i] = pW1[i];
  for (int i = tid; i < 64; i += 128)       sB1[i] = pb1[i];
  for (int i = tid; i < 64 * 128; i += 128) sW2[i] = pW2[i];
  if (tid < 128) sB2[tid] = pb2[tid];
  if (tid < 3) cent[tid] = 0.f;
  pool[tid] = 0.f;
  __syncthreads();

  const size_t pbase = ((size_t)obj * NPTS + tid) * 3;
  const float x = xyzs[pbase + 0], y = xyzs[pbase + 1], z = xyzs[pbase + 2];
  atomicAdd(&cent[0], x); atomicAdd(&cent[1], y); atomicAdd(&cent[2], z);
  __syncthreads();

  const float f0 = x - cent[0] * (1.f / NPTS);
  const float f1 = y - cent[1] * (1.f / NPTS);
  const float f2 = z - cent[2] * (1.f / NPTS);
  const float f3 = rgbs[pbase + 0], f4 = rgbs[pbase + 1], f5 = rgbs[pbase + 2];

  float h1[64];
  #pragma unroll
  for (int j = 0; j < 64; ++j) {
    float acc = sB1[j] + f0 * sW1[j] + f1 * sW1[64 + j] + f2 * sW1[128 + j]
              + f3 * sW1[192 + j] + f4 * sW1[256 + j] + f5 * sW1[320 + j];
    h1[j] = fmaxf(acc, 0.f);
  }

  for (int j = 0; j < 128; ++j) {
    const int jj = (j + tid) & 127;       // stagger to reduce LDS contention
    float acc = sB2[jj];
    #pragma unroll
    for (int i = 0; i < 64; ++i) acc += h1[i] * sW2[i * 128 + jj];
    acc = fmaxf(acc, 0.f);
    atomicMax((int*)&pool[jj], __float_as_int(acc));
  }
  __syncthreads();
  pooled[(size_t)obj * NPTS + tid] = pool[tid];
}

// ---------------------------------------------------------------------------
// LayerNorm (optional residual add, optional trailing relu); 1 wave per row.
// width <= 256. No __restrict__: used in-place (Out may alias Rsd).
// ---------------------------------------------------------------------------
__global__ void __launch_bounds__(128) ln_kernel(
    const float* X, const float* Rsd, const float* g, const float* bta,
    float* Out, int rows, int width, int relu) {
  const int row  = blockIdx.x * 4 + (threadIdx.x >> 5);
  const int lane = threadIdx.x & 31;
  if (row >= rows) return;
  const float* xp = X + (size_t)row * width;
  const float* rp = Rsd ? (Rsd + (size_t)row * width) : nullptr;

  float vals[8];
  float s = 0.f, s2 = 0.f;
  #pragma unroll
  for (int t = 0; t < 8; ++t) {
    const int c = lane + 32 * t;
    float v = 0.f;
    if (c < width) { v = xp[c]; if (rp) v += rp[c]; }
    vals[t] = v; s += v; s2 += v * v;
  }
  #pragma unroll
  for (int o = 16; o >= 1; o >>= 1) { s += __shfl_xor(s, o); s2 += __shfl_xor(s2, o); }
  const float m    = s / (float)width;
  const float var  = s2 / (float)width - m * m;
  const float rstd = rsqrtf(var + 1e-5f);
  float* op = Out + (size_t)row * width;
  #pragma unroll
  for (int t = 0; t < 8; ++t) {
    const int c = lane + 32 * t;
    if (c < width) {
      float v = (vals[t] - m) * rstd * g[c] + bta[c];
      if (relu) v = fmaxf(v, 0.f);
      op[c] = v;
    }
  }
}

// ---------------------------------------------------------------------------
// Sequence assembly: xseq[b,s,:] = [word|pos|tt] for s<16, [obj248|tt] else.
// ---------------------------------------------------------------------------
__global__ void embed_kernel(
    const float* __restrict__ obj248, const int* __restrict__ sentence,
    const int* __restrict__ tt_idx, const int* __restrict__ pos_idx,
    const float* __restrict__ word_emb, const float* __restrict__ tt_emb,
    const float* __restrict__ pos_emb, float* __restrict__ xseq) {
  const int idx = blockIdx.x * 256 + threadIdx.x;   // ROWS*256 exact
  const int d = idx & 255;
  const int r = idx >> 8;
  const int b = r / SLEN, s = r % SLEN;
  float v;
  if (s < WSENT) {
    if (d < 240)      v = word_emb[sentence[b * WSENT + s] * 240 + d];
    else if (d < 248) v = pos_emb[pos_idx[b * SLEN + s] * 8 + (d - 240)];
    else              v = tt_emb[tt_idx[b * SLEN + s] * 8 + (d - 248)];
  } else {
    const int k = s - WSENT;
    if (d < 248) v = obj248[(size_t)(b * KOBJ + k) * 248 + d];
    else         v = tt_emb[tt_idx[b * SLEN + s] * 8 + (d - 248)];
  }
  xseq[idx] = v;
}

// ---------------------------------------------------------------------------
// Attention: one wave per (b,h). S=28, hd=32; softmax with pad-mask bias.
// ---------------------------------------------------------------------------
__global__ void __launch_bounds__(32) attn_kernel(
    const float* __restrict__ Q, const float* __restrict__ Kb,
    const float* __restrict__ Vb, const int* __restrict__ sent_pad,
    const int* __restrict__ obj_pad, float* __restrict__ ctx) {
  const int bh = blockIdx.x;
  const int b = bh >> 3, h = bh & 7;
  const int lane = threadIdx.x;
  __shared__ float ks[SLEN][HDIM];
  __shared__ float vs[SLEN][HDIM];
  __shared__ float biasS[SLEN];

  for (int i = lane; i < SLEN * HDIM; i += 32) {
    const int s = i >> 5, d = i & 31;
    const size_t off = ((size_t)(b * SLEN + s)) * DMODEL + h * HDIM + d;
    ks[s][d] = Kb[off];
    vs[s][d] = Vb[off];
  }
  if (lane < SLEN) {
    const int pm = (lane < WSENT) ? sent_pad[b * WSENT + lane]
                                  : obj_pad[b * KOBJ + lane - WSENT];
    biasS[lane] = (pm == 1) ? -1e9f : 0.f;
  }
  __syncthreads();

  if (lane < SLEN) {
    float qv[HDIM];
    const float* qp = Q + ((size_t)(b * SLEN + lane)) * DMODEL + h * HDIM;
    #pragma unroll
    for (int d = 0; d < HDIM; ++d) qv[d] = qp[d];

    const float scale = 0.17677669529663687f;       // 1/sqrt(32)
    float sc[SLEN], mx = -1e30f;
    #pragma unroll
    for (int k = 0; k < SLEN; ++k) {
      float acc = 0.f;
      #pragma unroll
      for (int d = 0; d < HDIM; ++d) acc += qv[d] * ks[k][d];
      acc = acc * scale + biasS[k];
      sc[k] = acc; mx = fmaxf(mx, acc);
    }
    float den = 0.f;
    #pragma unroll
    for (int k = 0; k < SLEN; ++k) { sc[k] = expf(sc[k] - mx); den += sc[k]; }
    const float inv = 1.f / den;
    float out[HDIM];
    #pragma unroll
    for (int d = 0; d < HDIM; ++d) out[d] = 0.f;
    #pragma unroll
    for (int k = 0; k < SLEN; ++k) {
      const float p = sc[k] * inv;
      #pragma unroll
      for (int d = 0; d < HDIM; ++d) out[d] += p * vs[k][d];
    }
    float* op = ctx + ((size_t)(b * SLEN + lane)) * DMODEL + h * HDIM;
    #pragma unroll
    for (int d = 0; d < HDIM; ++d) op[d] = out[d];
  }
}

// ---------------------------------------------------------------------------
// Fused FF (256->16 relu ->256) + residual + LayerNorm; 1 wave per row.
// ---------------------------------------------------------------------------
__global__ void __launch_bounds__(128) ffn_ln_kernel(
    float* xseq, const float* __restrict__ W1, const float* __restrict__ b1,
    const float* __restrict__ W2, const float* __restrict__ b2,
    const float* __restrict__ g, const float* __restrict__ bta) {
  __shared__ float xs[4][DMODEL];
  __shared__ float hs[4][FFDIM];
  const int wave = threadIdx.x >> 5, lane = threadIdx.x & 31;
  const int row = blockIdx.x * 4 + wave;
  float* xp = xseq + (size_t)row * DMODEL;

  #pragma unroll
  for (int t = 0; t < 8; ++t) xs[wave][lane + 32 * t] = xp[lane + 32 * t];
  __syncthreads();

  if (lane < FFDIM) {
    float acc = b1[lane];
    for (int i = 0; i < DMODEL; ++i) acc += xs[wave][i] * W1[i * FFDIM + lane];
    hs[wave][lane] = fmaxf(acc, 0.f);
  }
  __syncthreads();

  float vals[8];
  float s = 0.f, s2 = 0.f;
  #pragma unroll
  for (int t = 0; t < 8; ++t) {
    const int c = lane + 32 * t;
    float acc = b2[c];
    #pragma unroll
    for (int j = 0; j < FFDIM; ++j) acc += hs[wave][j] * W2[j * DMODEL + c];
    const float v = xs[wave][c] + acc;
    vals[t] = v; s += v; s2 += v * v;
  }
  #pragma unroll
  for (int o = 16; o >= 1; o >>= 1) { s += __shfl_xor(s, o); s2 += __shfl_xor(s2, o); }
  const float m = s * (1.f / DMODEL);
  const float rstd = rsqrtf(s2 * (1.f / DMODEL) - m * m + 1e-5f);
  #pragma unroll
  for (int t = 0; t < 8; ++t) {
    const int c = lane + 32 * t;
    xp[c] = (vals[t] - m) * rstd * g[c] + bta[c];
  }
}

// ---------------------------------------------------------------------------
// Gather object rows out of xseq.
// ---------------------------------------------------------------------------
__global__ void gather_obj_kernel(const float* __restrict__ xseq,
                                  float* __restrict__ objenc) {
  const int idx = blockIdx.x * 256 + threadIdx.x;   // BK*256 exact
  const int d = idx & 255;
  const int r = idx >> 8;
  const int b = r / KOBJ, k = r % KOBJ;
  objenc[idx] = xseq[((size_t)(b * SLEN + WSENT + k)) * DMODEL + d];
}

// ---------------------------------------------------------------------------
// Gumbel-Sinkhorn: one wave per 12x12 matrix (4 per block). Lane r owns row
// r in registers; column logsumexp via LDS. Emits P (transposed) + seq_pred.
// ---------------------------------------------------------------------------
__global__ void __launch_bounds__(128) sinkhorn_kernel(
    const float* __restrict__ log_alpha, const float* __restrict__ nsamp,
    const int* __restrict__ pos_idx, float* __restrict__ seq_pred,
    float* __restrict__ Pout) {
  __shared__ float sm[4][13][KOBJ];
  const int wave = threadIdx.x >> 5, lane = threadIdx.x & 31;
  const int n = blockIdx.x * 4 + wave;          // 0..2559
  const int b = n & (B_ - 1);                   // n % 512

  float la[KOBJ];
  if (lane < KOBJ) {
    const float* lap = log_alpha + (size_t)b * (KOBJ * KOBJ) + lane * KOBJ;
    const float* nsp = nsamp + (size_t)n * (KOBJ * KOBJ) + lane * KOBJ;
    #pragma unroll
    for (int c = 0; c < KOBJ; ++c)
      la[c] = lap[c] - logf(1e-20f - logf(nsp[c] + 1e-20f));
  }

  for (int it = 0; it < NITER; ++it) {
    if (lane < KOBJ) {                          // row logsumexp (axis 2)
      float mx = -1e30f;
      #pragma unroll
      for (int c = 0; c < KOBJ; ++c) mx = fmaxf(mx, la[c]);
      float ssum = 0.f;
      #pragma unroll
      for (int c = 0; c < KOBJ; ++c) ssum += expf(la[c] - mx);
      const float lse = mx + logf(ssum);
      #pragma unroll
      for (int c = 0; c < KOBJ; ++c) { la[c] -= lse; sm[wave][lane][c] = la[c]; }
    }
    __syncthreads();
    if (lane < KOBJ) {                          // column logsumexp (axis 1)
      float mx = -1e30f;
      #pragma unroll
      for (int r = 0; r < KOBJ; ++r) mx = fmaxf(mx, sm[wave][r][lane]);
      float ssum = 0.f;
      #pragma unroll
      for (int r = 0; r < KOBJ; ++r) ssum += expf(sm[wave][r][lane] - mx);
      sm[wave][12][lane] = mx + logf(ssum);
    }
    __syncthreads();
    if (lane < KOBJ) {
      #pragma unroll
      for (int c = 0; c < KOBJ; ++c) la[c] -= sm[wave][12][c];
    }
    __syncthreads();
  }

  if (lane < KOBJ) {
    #pragma unroll
    for (int c = 0; c < KOBJ; ++c) sm[wave][lane][c] = la[c];
  }
  __syncthreads();
  if (lane < KOBJ) {
    // P[n, lane, r] = exp(la[n, r, lane]);  seq_pred[n, lane] = P @ seqf
    float acc = 0.f;
    #pragma unroll
    for (int r = 0; r < KOBJ; ++r) {
      const float p = expf(sm[wave][r][lane]);
      Pout[(size_t)n * (KOBJ * KOBJ) + lane * KOBJ + r] = p;
      acc += p * (float)pos_idx[b * SLEN + WSENT + r];
    }
    seq_pred[n * KOBJ + lane] = acc;
  }
}

// ---------------------------------------------------------------------------
// Host launch
// ---------------------------------------------------------------------------
static const int kParamSizes[37] = {
  384, 64, 8192, 128, 32768, 256,            // pW1 pb1 pW2 pb2 pWp pbp
  126976, 496, 123008, 248, 248, 248,        // fc1 fb1 fc2 fb2 mg mb
  240000, 16, 96,                            // word_emb tt_emb pos_emb
  524288, 2048, 524288, 2048, 524288, 2048, 524288, 2048,  // Wq bq Wk bk Wv bv Wo bo
  2048, 2048, 2048, 2048,                    // ln1g ln1b ln2g ln2b
  32768, 128, 32768, 2048,                   // W1 b1 W2 b2
  65536, 256, 256, 256, 3072, 12             // sW1 sb1 sg sb sW2 sb2
};
enum { iPW1, iPB1, iPW2, iPB2, iPWP, iPBP, iFC1, iFB1, iFC2, iFB2, iMG, iMB,
       iWORD, iTT, iPOS, iWQ, iBQ, iWK, iBK, iWV, iBV, iWO, iBO,
       iLN1G, iLN1B, iLN2G, iLN2B, iW1, iB1, iW2, iB2,
       iSW1, iSB1, iSG, iSB, iSW2, iSB2 };

extern "C" void kernel_launch(void* const* d_in, const int* in_sizes, int n_in,
                              void* d_out, int out_size, void* d_ws, size_t ws_size,
                              hipStream_t stream) {
  const float* xyzs     = (const float*)d_in[0];
  const float* rgbs     = (const float*)d_in[1];
  const int*   obj_pad  = (const int*)d_in[2];
  const int*   sentence = (const int*)d_in[3];
  const int*   sent_pad = (const int*)d_in[4];
  const int*   tt_idx   = (const int*)d_in[5];
  const int*   pos_idx  = (const int*)d_in[6];
  const float* nsamp    = (const float*)d_in[7];

  const float* P[37];
  if (n_in >= 45) {
    for (int i = 0; i < 37; ++i) P[i] = (const float*)d_in[8 + i];
  } else {                                   // single concatenated params blob
    const float* base = (const float*)d_in[8];
    size_t off = 0;
    for (int i = 0; i < 37; ++i) { P[i] = base + off; off += kParamSizes[i]; }
  }

  float* ws = (float*)d_ws;
  const size_t RB = (size_t)ROWS * DMODEL;   // 3,670,016 floats
  float* X  = ws;
  float* T0 = ws + RB;
  float* T1 = T0 + RB;
  float* T2 = T1 + RB;
  float* T3 = T2 + RB;
  float* T4 = T3 + RB;

  auto gemm = [&](const float* A, const float* Wt, const float* bias, float* C,
                  int M, int Kd, int Nout, bool relu) {
    dim3 grid((Nout + 63) / 64, M / 64);
    if (relu)
      gemm_bias_kernel<true><<<grid, 128, 0, stream>>>(A, Wt, bias, C, M, Kd, Nout);
    else
      gemm_bias_kernel<false><<<grid, 128, 0, stream>>>(A, Wt, bias, C, M, Kd, Nout);
  };

  // ---- Stage 1: point encoder + per-object MLP -> obj248 ----
  point_enc_kernel<<<BK, 128, 0, stream>>>(xyzs, rgbs, P[iPW1], P[iPB1],
                                           P[iPW2], P[iPB2], T0);
  gemm(T0, P[iPWP], P[iPBP], T1, BK, 128, 256, false);       // xfeat
  gemm(T1, P[iFC1], P[iFB1], T2, BK, 256, 496, true);        // relu(fc1)
  gemm(T2, P[iFC2], P[iFB2], T3, BK, 496, 248, false);       // fc2
  ln_kernel<<<BK / 4, 128, 0, stream>>>(T3, nullptr, P[iMG], P[iMB], T4,
                                        BK, 248, 0);          // obj248

  // ---- Stage 2: sequence assembly ----
  embed_kernel<<<ROWS, 256, 0, stream>>>(T4, sentence, tt_idx, pos_idx,
                                         P[iWORD], P[iTT], P[iPOS], X);

  // ---- Stage 3: transformer layers ----
  for (int l = 0; l < NLAYER; ++l) {
    const size_t wOff = (size_t)l * DMODEL * DMODEL;
    const size_t bOff = (size_t)l * DMODEL;
    gemm(X, P[iWQ] + wOff, P[iBQ] + bOff, T0, ROWS, DMODEL, DMODEL, false);
    gemm(X, P[iWK] + wOff, P[iBK] + bOff, T1, ROWS, DMODEL, DMODEL, false);
    gemm(X, P[iWV] + wOff, P[iBV] + bOff, T2, ROWS, DMODEL, DMODEL, false);
    attn_kernel<<<B_ * NHEAD, 32, 0, stream>>>(T0, T1, T2, sent_pad, obj_pad, T3);
    gemm(T3, P[iWO] + wOff, P[iBO] + bOff, T4, ROWS, DMODEL, DMODEL, false);
    ln_kernel<<<ROWS / 4, 128, 0, stream>>>(T4, X, P[iLN1G] + bOff,
                                            P[iLN1B] + bOff, X, ROWS, DMODEL, 0);
    ffn_ln_kernel<<<ROWS / 4, 128, 0, stream>>>(
        X, P[iW1] + (size_t)l * DMODEL * FFDIM, P[iB1] + (size_t)l * FFDIM,
        P[iW2] + (size_t)l * FFDIM * DMODEL, P[iB2] + bOff,
        P[iLN2G] + bOff, P[iLN2B] + bOff);
  }

  // ---- Stage 4: head -> log_alpha ----
  gather_obj_kernel<<<BK, 256, 0, stream>>>(X, T0);
  gemm(T0, P[iSW1], P[iSB1], T1, BK, DMODEL, DMODEL, false);
  ln_kernel<<<BK / 4, 128, 0, stream>>>(T1, nullptr, P[iSG], P[iSB], T2,
                                        BK, DMODEL, 1);       // relu(LN(.))
  gemm(T2, P[iSW2], P[iSB2], T3, BK, DMODEL, KOBJ, false);    // log_alpha

  // ---- Stage 5: Gumbel-Sinkhorn -> (seq_pred, P) in d_out ----
  float* outp = (float*)d_out;
  float* seq_pred = outp;                         // 2560 * 12
  float* Pmat = outp + (size_t)NSAMP * B_ * KOBJ; // 2560 * 144
  sinkhorn_kernel<<<(NSAMP * B_) / 4, 128, 0, stream>>>(T3, nsamp, pos_idx,
                                                        seq_pred, Pmat);
}